// GRANDE_72619307041123
// MI455X (gfx1250) — compile-verified
//
#include <hip/hip_runtime.h>

typedef __attribute__((ext_vector_type(2))) float v2f;
typedef __attribute__((ext_vector_type(8))) float v8f;

#define KDIM 256
#define WPB  8   // waves (== rows) per block

// One wave32 per row of 256. 8 elements per lane (e = lane*8 + r).
// Pipeline: vectorized load -> in-register bitonic sort (shfl_xor for
// strides >= 8) -> LDS stage of shifted sorted values -> cumsum and
// cumsum-of-squares via chained v_wmma_f32_16x16x4_f32 (triangular-ones
// matmuls) -> tau candidates / support count -> relu(x - tau*)^2 store.
__global__ __launch_bounds__(256) void entmax15_grande(const float* __restrict__ in,
                                                       float* __restrict__ out,
                                                       int nrows) {
  const int lane = threadIdx.x & 31;
  const int wave = threadIdx.x >> 5;
  const int row  = blockIdx.x * WPB + wave;
  const int rowc = (row < nrows) ? row : (nrows - 1);   // clamp: keep flow uniform for WMMA

  __shared__ float lds[WPB][KDIM + 32];
  float* zbuf = lds[wave];          // 256 sorted/shifted values, later overwritten with tau
  float* seg1 = zbuf + KDIM;        // 16 segment sums of z
  float* seg2 = zbuf + KDIM + 16;   // 16 segment sums of z^2

  // ---- load 8 contiguous f32 per lane (two 128-bit loads), scale by 0.5
  const float4* src = reinterpret_cast<const float4*>(in + (size_t)rowc * KDIM);
  float4 q0 = src[lane * 2 + 0];
  float4 q1 = src[lane * 2 + 1];
  float h[8] = {q0.x, q0.y, q0.z, q0.w, q1.x, q1.y, q1.z, q1.w};
  float s[8];
#pragma unroll
  for (int r = 0; r < 8; ++r) { h[r] *= 0.5f; s[r] = h[r]; }

  // ---- bitonic sort, descending over 256 elements; element index e = lane*8 + r.
  // Strides 1,2,4 are intra-lane register exchanges; strides 8..128 use shfl_xor.
#pragma unroll
  for (int k = 2; k <= 256; k <<= 1) {
#pragma unroll
    for (int j = k >> 1; j > 0; j >>= 1) {
      if (j >= 8) {
        const int lm = j >> 3;
        const bool ilow = ((lane & lm) == 0);
#pragma unroll
        for (int r = 0; r < 8; ++r) {
          float other = __shfl_xor(s[r], lm, 32);
          const int i = (lane << 3) | r;
          const bool desc = ((i & k) == 0);
          float mx = fmaxf(s[r], other), mn = fminf(s[r], other);
          s[r] = (ilow == desc) ? mx : mn;
        }
      } else {
#pragma unroll
        for (int r = 0; r < 8; ++r) {
          if ((r & j) == 0) {
            const int r2 = r | j;
            const int i = (lane << 3) | r;
            const bool desc = ((i & k) == 0);
            float a = s[r], b = s[r2];
            float mx = fmaxf(a, b), mn = fminf(a, b);
            s[r]  = desc ? mx : mn;
            s[r2] = desc ? mn : mx;
          }
        }
      }
    }
  }

  // max of the row is the first sorted element (sort order is shift-invariant)
  const float maxv = __shfl(s[0], 0, 32);

  // stage shifted sorted values z[e] = xs[e] - max to LDS
#pragma unroll
  for (int r = 0; r < 8; ++r) zbuf[(lane << 3) | r] = s[r] - maxv;
  __syncthreads();

  // 16 segment sums (and sums of squares) of length-16 segments
  if (lane < 16) {
    float a1 = 0.f, a2 = 0.f;
#pragma unroll
    for (int jj = 0; jj < 16; ++jj) {
      float zz = zbuf[lane * 16 + jj];
      a1 += zz; a2 += zz * zz;
    }
    seg1[lane] = a1;
    seg2[lane] = a2;
  }
  __syncthreads();

  // ---- cumsum(z) and cumsum(z^2) via WMMA.
  // Z reshaped 16x16 (Z[i][j] = z[16i+j]). cumsum = Lstrict @ Sbcast + Z @ U,
  // where Lstrict[i][k] = (k < i), U[j][n] = (j <= n), Sbcast[k][n] = segsum[k].
  // f32 fragment striping: A rows / B cols / C cols indexed by (lane & 15);
  // lanes >= 16 carry the second half of the contraction (A/B) or rows 8..15 (C).
  const int m = lane & 15;
  const int half = lane >> 4;

  v8f c1 = {};
  v8f c2 = {};

#pragma unroll
  for (int kk = 0; kk < 4; ++kk) {          // phase 1: Lstrict @ Sbcast
    const int k0 = 4 * kk + 2 * half;
    v2f a  = { (k0 < m) ? 1.0f : 0.0f, (k0 + 1 < m) ? 1.0f : 0.0f };
    v2f b1 = { seg1[k0], seg1[k0 + 1] };
    v2f b2 = { seg2[k0], seg2[k0 + 1] };
    c1 = __builtin_amdgcn_wmma_f32_16x16x4_f32(false, a, false, b1, (short)0, c1, false, false);
    c2 = __builtin_amdgcn_wmma_f32_16x16x4_f32(false, a, false, b2, (short)0, c2, false, false);
  }
#pragma unroll
  for (int kk = 0; kk < 4; ++kk) {          // phase 2: Z @ U (within-segment scan)
    const int k0 = 4 * kk + 2 * half;
    float z0 = zbuf[16 * m + k0];
    float z1 = zbuf[16 * m + k0 + 1];
    v2f a1v = { z0, z1 };
    v2f a2v = { z0 * z0, z1 * z1 };
    v2f bu  = { (k0 <= m) ? 1.0f : 0.0f, (k0 + 1 <= m) ? 1.0f : 0.0f };
    c1 = __builtin_amdgcn_wmma_f32_16x16x4_f32(false, a1v, false, bu, (short)0, c1, false, false);
    c2 = __builtin_amdgcn_wmma_f32_16x16x4_f32(false, a2v, false, bu, (short)0, c2, false, false);
  }

  // ---- tau candidates, support count; overwrite zbuf with tau for the final pick
  int cnt = 0;
#pragma unroll
  for (int v = 0; v < 8; ++v) {
    const int e = 16 * (v + 8 * half) + m;  // C-fragment element index
    float rho  = (float)(e + 1);
    float mean = c1[v] / rho;
    float msq  = c2[v] / rho;
    float delta = (1.0f - rho * (msq - mean * mean)) / rho;
    float tau = mean - sqrtf(fmaxf(delta, 0.0f));
    float ze = zbuf[e];
    cnt += (tau <= ze) ? 1 : 0;
    zbuf[e] = tau;
  }
  __syncthreads();
#pragma unroll
  for (int off = 16; off > 0; off >>= 1) cnt += __shfl_xor(cnt, off, 32);

  const float tau_star = zbuf[cnt - 1];
  const float t = maxv + tau_star;          // y = relu((x/2 - max) - tau*) = relu(h - t)

  if (row < nrows) {
    float4* dst = reinterpret_cast<float4*>(out + (size_t)row * KDIM);
    float y[8];
#pragma unroll
    for (int r = 0; r < 8; ++r) { float u = fmaxf(h[r] - t, 0.0f); y[r] = u * u; }
    dst[lane * 2 + 0] = make_float4(y[0], y[1], y[2], y[3]);
    dst[lane * 2 + 1] = make_float4(y[4], y[5], y[6], y[7]);
  }
}

extern "C" void kernel_launch(void* const* d_in, const int* in_sizes, int n_in,
                              void* d_out, int out_size, void* d_ws, size_t ws_size,
                              hipStream_t stream) {
  (void)n_in; (void)out_size; (void)d_ws; (void)ws_size;
  const float* logits = (const float*)d_in[0];
  float* out = (float*)d_out;
  const int nrows = in_sizes[0] / KDIM;     // 2048 * 127 = 260096 rows of 256
  dim3 grid((nrows + WPB - 1) / WPB);
  dim3 block(256);
  hipLaunchKernelGGL(entmax15_grande, grid, block, 0, stream, logits, out, nrows);
}